// GCNNet_37177236914410
// MI455X (gfx1250) — compile-verified
//
#include <hip/hip_runtime.h>

typedef __attribute__((ext_vector_type(2))) float v2f;
typedef __attribute__((ext_vector_type(8))) float v8f;

// ---------------------------------------------------------------------------
// Degree / normalization: deg[i] = 1 (self loop) + #edges with dst==i,
// then dinv[i] = rsqrt(deg[i])  (deg >= 1 always, so no zero guard needed)
// ---------------------------------------------------------------------------
__global__ void __launch_bounds__(256) k_init_deg(float* __restrict__ deg, int N) {
  int i = blockIdx.x * blockDim.x + threadIdx.x;
  if (i < N) deg[i] = 1.0f;
}

__global__ void __launch_bounds__(256) k_count_deg(const long long* __restrict__ dst,
                                                   float* __restrict__ deg, int E) {
  int e = blockIdx.x * blockDim.x + threadIdx.x;
  if (e < E) atomicAdd(&deg[(int)dst[e]], 1.0f);
}

__global__ void __launch_bounds__(256) k_dinv(float* __restrict__ deg, int N) {
  int i = blockIdx.x * blockDim.x + threadIdx.x;
  if (i < N) deg[i] = rsqrtf(deg[i]);
}

// ---------------------------------------------------------------------------
// GEMM1 via f32 WMMA: H[N,64] = X[N,128] @ W1[128,64]
// Epilogue also seeds AGG with the self-loop message: AGG = H * dinv^2.
// One block = 128 threads = 4 waves, each wave owns one 16-wide column tile.
// f32 WMMA operand layout (ISA 7.12.2):
//   A 16x4:  lanes 0-15 -> M=lane, {v0,v1} = K{0,1}; lanes 16-31 -> K{2,3}
//   B 4x16:  lanes 0-15 -> N=lane, {v0,v1} = K{0,1}; lanes 16-31 -> K{2,3}
//   C 16x16: vgpr i -> row i (lanes 0-15) / row i+8 (lanes 16-31), col = lane&15
// ---------------------------------------------------------------------------
__global__ void __launch_bounds__(128) k_gemm1(const float* __restrict__ X,
                                               const float* __restrict__ W1,
                                               const float* __restrict__ dinv,
                                               float* __restrict__ H,
                                               float* __restrict__ AGG, int N) {
  const int lane = threadIdx.x & 31;
  const int col0 = (threadIdx.x >> 5) << 4;    // 0,16,32,48
  const int row0 = blockIdx.x << 4;
  const int mr   = lane & 15;
  const int kh   = (lane >> 4) << 1;           // 0 or 2
  const int ar   = min(row0 + mr, N - 1);      // clamp for tail tile
  const float* xr = X + (size_t)ar * 128;

  v8f c = {};
#pragma unroll
  for (int k = 0; k < 128; k += 4) {
    v2f a, b;
    a.x = xr[k + kh];
    a.y = xr[k + kh + 1];
    b.x = W1[(size_t)(k + kh)     * 64 + col0 + mr];
    b.y = W1[(size_t)(k + kh + 1) * 64 + col0 + mr];
    c = __builtin_amdgcn_wmma_f32_16x16x4_f32(false, a, false, b, (short)0, c,
                                              false, false);
  }

  const int rbase = row0 + ((lane >> 4) << 3);
#pragma unroll
  for (int i = 0; i < 8; ++i) {
    int r = rbase + i;
    if (r < N) {
      float di = dinv[r];
      float v  = c[i];
      size_t o = (size_t)r * 64 + col0 + mr;
      H[o]   = v;
      AGG[o] = v * (di * di);
    }
  }
}

// ---------------------------------------------------------------------------
// Edge aggregation layer 1: AGG[dst] += H[src] * dinv[src]*dinv[dst]
// One wave per edge; each lane handles 2 of the 64 columns (fully coalesced
// 128B gather + 128B atomic scatter per half-row). Working set (~52MB) fits
// the 192MB L2, so the f32 atomics mostly resolve in L2.
// ---------------------------------------------------------------------------
__global__ void __launch_bounds__(256) k_scatter1(const long long* __restrict__ src,
                                                  const long long* __restrict__ dst,
                                                  const float* __restrict__ H,
                                                  const float* __restrict__ dinv,
                                                  float* __restrict__ AGG, int E) {
  int e = blockIdx.x * 8 + (threadIdx.x >> 5);
  if (e >= E) return;                           // wave-uniform exit
  int lane = threadIdx.x & 31;
  int s = (int)src[e], d = (int)dst[e];
  float nrm = dinv[s] * dinv[d];
  size_t so = (size_t)s * 64 + lane;
  size_t dofs = (size_t)d * 64 + lane;
  atomicAdd(&AGG[dofs],      H[so]      * nrm);
  atomicAdd(&AGG[dofs + 32], H[so + 32] * nrm);
}

// ---------------------------------------------------------------------------
// GEMM2 via f32 WMMA with fused pre/post ops:
//   A-load:   a = relu(AGG + b1)           (layer-1 bias + ReLU)
//   epilogue: H2 = a@W2 ; OUT = b2 + H2*dinv^2  (self-loop seed + bias)
// One block = 64 threads = 2 waves (two 16-wide column tiles of the 32 cols).
// ---------------------------------------------------------------------------
__global__ void __launch_bounds__(64) k_gemm2(const float* __restrict__ AGG,
                                              const float* __restrict__ W2,
                                              const float* __restrict__ b1,
                                              const float* __restrict__ b2,
                                              const float* __restrict__ dinv,
                                              float* __restrict__ H2,
                                              float* __restrict__ OUT, int N) {
  const int lane = threadIdx.x & 31;
  const int col0 = (threadIdx.x >> 5) << 4;    // 0 or 16
  const int row0 = blockIdx.x << 4;
  const int mr   = lane & 15;
  const int kh   = (lane >> 4) << 1;
  const int ar   = min(row0 + mr, N - 1);
  const float* arow = AGG + (size_t)ar * 64;

  v8f c = {};
#pragma unroll
  for (int k = 0; k < 64; k += 4) {
    v2f a, b;
    a.x = fmaxf(arow[k + kh]     + b1[k + kh],     0.0f);
    a.y = fmaxf(arow[k + kh + 1] + b1[k + kh + 1], 0.0f);
    b.x = W2[(size_t)(k + kh)     * 32 + col0 + mr];
    b.y = W2[(size_t)(k + kh + 1) * 32 + col0 + mr];
    c = __builtin_amdgcn_wmma_f32_16x16x4_f32(false, a, false, b, (short)0, c,
                                              false, false);
  }

  const float bc = b2[col0 + mr];
  const int rbase = row0 + ((lane >> 4) << 3);
#pragma unroll
  for (int i = 0; i < 8; ++i) {
    int r = rbase + i;
    if (r < N) {
      float di = dinv[r];
      float v  = c[i];
      size_t o = (size_t)r * 32 + col0 + mr;
      H2[o]  = v;
      OUT[o] = bc + v * (di * di);
    }
  }
}

// ---------------------------------------------------------------------------
// Edge aggregation layer 2: OUT[dst] += H2[src] * dinv[src]*dinv[dst]
// One wave per edge, one lane per column (32 cols, 128B coalesced).
// ---------------------------------------------------------------------------
__global__ void __launch_bounds__(256) k_scatter2(const long long* __restrict__ src,
                                                  const long long* __restrict__ dst,
                                                  const float* __restrict__ H2,
                                                  const float* __restrict__ dinv,
                                                  float* __restrict__ OUT, int E) {
  int e = blockIdx.x * 8 + (threadIdx.x >> 5);
  if (e >= E) return;
  int lane = threadIdx.x & 31;
  int s = (int)src[e], d = (int)dst[e];
  float nrm = dinv[s] * dinv[d];
  atomicAdd(&OUT[(size_t)d * 32 + lane], H2[(size_t)s * 32 + lane] * nrm);
}

// ---------------------------------------------------------------------------
// Host launcher
// Inputs: x[N,128] f32, edge_index[2,E] i64, W1[128,64], b1[64], W2[64,32], b2[32]
// Output: [N,32] f32
// Workspace: dinv (N f32) | H (N*64 f32, reused as H2) | AGG (N*64 f32) ~= 52MB
// ---------------------------------------------------------------------------
extern "C" void kernel_launch(void* const* d_in, const int* in_sizes, int n_in,
                              void* d_out, int out_size, void* d_ws, size_t ws_size,
                              hipStream_t stream) {
  const float*     x   = (const float*)d_in[0];
  const long long* ei  = (const long long*)d_in[1];
  const float*     W1  = (const float*)d_in[2];
  const float*     b1  = (const float*)d_in[3];
  const float*     W2  = (const float*)d_in[4];
  const float*     b2  = (const float*)d_in[5];
  float*           out = (float*)d_out;

  const int N = in_sizes[0] / 128;
  const int E = in_sizes[1] / 2;
  const long long* src = ei;        // edge_index[0]
  const long long* dst = ei + E;    // edge_index[1]

  float* ws   = (float*)d_ws;
  float* dinv = ws;                               // N floats
  float* H    = ws + 131072;                      // N*64 floats (also H2)
  float* AGG  = H + (size_t)N * 64;               // N*64 floats
  float* H2   = H;                                // reuse: H dead after scatter1

  const int rowTiles = (N + 15) / 16;
  const int edgeBlks = (E + 7) / 8;

  k_init_deg <<<(N + 255) / 256, 256, 0, stream>>>(dinv, N);
  k_count_deg<<<(E + 255) / 256, 256, 0, stream>>>(dst, dinv, E);
  k_dinv     <<<(N + 255) / 256, 256, 0, stream>>>(dinv, N);

  k_gemm1    <<<rowTiles, 128, 0, stream>>>(x, W1, dinv, H, AGG, N);
  k_scatter1 <<<edgeBlks, 256, 0, stream>>>(src, dst, H, dinv, AGG, E);
  k_gemm2    <<<rowTiles,  64, 0, stream>>>(AGG, W2, b1, b2, dinv, H2, out, N);
  k_scatter2 <<<edgeBlks, 256, 0, stream>>>(src, dst, H2, dinv, out, E);
}